// PaiNNInteraction_59141699666425
// MI455X (gfx1250) — compile-verified
//
#include <hip/hip_runtime.h>
#include <math.h>

// ---------------------------------------------------------------------------
// PaiNN interaction layer for MI455X (gfx1250, wave32).
//   Pass 1: node MLP  S = silu(ns@W1+b1)@W2+b2   via v_wmma_f32_16x16x4_f32
//   Pass 2: init d_out with residual node states
//   Pass 3: per-edge fused filter + gather + gate + atomic scatter-add
// ---------------------------------------------------------------------------

#define NUM_NODES 20000
#define NUM_EDGES 320000
#define NODE_SIZE 128
#define EDGE_SIZE 20
#define CUTOFF_F  5.0f
#define PI_F      3.14159265358979323846f

typedef __attribute__((ext_vector_type(2))) float v2f;
typedef __attribute__((ext_vector_type(8))) float v8f;

// One wave computes a 16x16 f32 tile of A(16xK) @ B(KxN-major ldb), cols n0..n0+15.
// A-layout (16x4 f32): lane l holds row m=l&15, K elements {k+2*(l>>4), +1} -> contiguous pair.
// B-layout (4x16 f32): lane l holds col n=l&15, same two K rows -> two strided loads.
__device__ __forceinline__ v8f wmma_gemm_tile_f32(const float* __restrict__ A, int lda,
                                                  const float* __restrict__ B, int ldb,
                                                  int n0, int K, int lane) {
  const int m    = lane & 15;
  const int half = lane >> 4;                 // 0: K pair {0,1}; 1: K pair {2,3}
  const int n    = n0 + (lane & 15);
  v8f acc = {0.f, 0.f, 0.f, 0.f, 0.f, 0.f, 0.f, 0.f};
  for (int k = 0; k < K; k += 4) {
    const int ka = k + 2 * half;
    v2f a;
    a.x = A[m * lda + ka];
    a.y = A[m * lda + ka + 1];
    v2f b;
    b.x = B[ka * ldb + n];
    b.y = B[(ka + 1) * ldb + n];
    acc = __builtin_amdgcn_wmma_f32_16x16x4_f32(
        /*neg_a=*/false, a, /*neg_b=*/false, b,
        /*c_mod=*/(short)0, acc, /*reuse_a=*/false, /*reuse_b=*/false);
  }
  return acc;
}

// ---------------------------------------------------------------------------
// Pass 1: per 16-node tile: S = silu(X@W1+b1)@W2+b2.  8 waves / block.
// ---------------------------------------------------------------------------
__global__ __launch_bounds__(256) void painn_node_mlp_kernel(
    const float* __restrict__ ns,   // (NUM_NODES, 128)
    const float* __restrict__ W1,   // (128, 128) row-major
    const float* __restrict__ b1,   // (128,)
    const float* __restrict__ W2,   // (128, 384) row-major
    const float* __restrict__ b2,   // (384,)
    float* __restrict__ S) {        // (NUM_NODES, 384)
  __shared__ float Xs[16 * NODE_SIZE];
  __shared__ float Hs[16 * NODE_SIZE];

  const int tile = blockIdx.x;                 // 20000/16 = 1250 tiles, exact
  const int tid  = threadIdx.x;
  const int wave = tid >> 5;                   // 0..7
  const int lane = tid & 31;

  // Stage X tile (16x128) into LDS.
  const float* __restrict__ xsrc = ns + (size_t)tile * 16 * NODE_SIZE;
  for (int i = tid; i < 16 * NODE_SIZE; i += 256) Xs[i] = xsrc[i];
  __syncthreads();

  // H = silu(X @ W1 + b1): each wave owns one 16-col tile of the 128 cols.
  {
    const int n0 = wave * 16;
    v8f acc = wmma_gemm_tile_f32(Xs, NODE_SIZE, W1, NODE_SIZE, n0, NODE_SIZE, lane);
    const int half = lane >> 4;
    const int n    = n0 + (lane & 15);
    const float bias = b1[n];
#pragma unroll
    for (int r = 0; r < 8; ++r) {
      const int mrow = r + 8 * half;           // D row per ISA C/D layout
      float x = acc[r] + bias;
      Hs[mrow * NODE_SIZE + n] = x / (1.0f + __expf(-x));   // silu
    }
  }
  __syncthreads();

  // S = H @ W2 + b2: 24 col tiles, 3 per wave.
  float* __restrict__ sdst = S + (size_t)tile * 16 * (3 * NODE_SIZE);
#pragma unroll
  for (int t = 0; t < 3; ++t) {
    const int n0 = (wave * 3 + t) * 16;
    v8f acc = wmma_gemm_tile_f32(Hs, NODE_SIZE, W2, 3 * NODE_SIZE, n0, NODE_SIZE, lane);
    const int half = lane >> 4;
    const int n    = n0 + (lane & 15);
    const float bias = b2[n];
#pragma unroll
    for (int r = 0; r < 8; ++r) {
      const int mrow = r + 8 * half;
      sdst[mrow * (3 * NODE_SIZE) + n] = acc[r] + bias;
    }
  }
}

// ---------------------------------------------------------------------------
// Pass 2: d_out = [node_state_scalar ; node_state_vector] (residual base).
// ---------------------------------------------------------------------------
__global__ __launch_bounds__(256) void painn_init_out_kernel(
    const float* __restrict__ ns, const float* __restrict__ nsv,
    float* __restrict__ out) {
  const size_t n_scalar = (size_t)NUM_NODES * NODE_SIZE;
  const size_t total    = n_scalar + (size_t)NUM_NODES * 3 * NODE_SIZE;
  for (size_t i = (size_t)blockIdx.x * blockDim.x + threadIdx.x; i < total;
       i += (size_t)gridDim.x * blockDim.x) {
    out[i] = (i < n_scalar) ? ns[i] : nsv[i - n_scalar];
  }
}

// ---------------------------------------------------------------------------
// Pass 3: one 128-thread block per edge; thread t owns channel t of each of
// the three gate groups. Filter GEMM (K=20) done inline (60 FMAs/thread).
// ---------------------------------------------------------------------------
__global__ __launch_bounds__(128) void painn_edge_kernel(
    const float* __restrict__ edge_state,     // (E, 20)
    const float* __restrict__ edge_vector,    // (E, 3)
    const float* __restrict__ edge_distance,  // (E, 1)
    const int*   __restrict__ edges,          // (E, 2)
    const float* __restrict__ Wf,             // (20, 384)
    const float* __restrict__ bf,             // (384,)
    const float* __restrict__ nsv,            // (N, 3, 128)
    const float* __restrict__ S,              // (N, 384)
    float* __restrict__ out_scalar,           // (N, 128) accumulators
    float* __restrict__ out_vector) {         // (N, 3, 128) accumulators
  const int e = blockIdx.x;
  const int t = threadIdx.x;                   // 0..127

  __shared__ float es[EDGE_SIZE];
  if (t < EDGE_SIZE) es[t] = edge_state[(size_t)e * EDGE_SIZE + t];
  __syncthreads();

  const int src = edges[2 * e];
  const int dst = edges[2 * e + 1];

  // cosine cutoff (scalar per edge; uniform compute per thread is cheap)
  const float dist = edge_distance[e];
  const float cc = (dist < CUTOFF_F)
                       ? 0.5f * (__cosf(dist * (PI_F / CUTOFF_F)) + 1.0f)
                       : 0.0f;

  // normalized edge vector
  const float ev0 = edge_vector[3 * e + 0];
  const float ev1 = edge_vector[3 * e + 1];
  const float ev2 = edge_vector[3 * e + 2];
  const float nrm = fmaxf(sqrtf(ev0 * ev0 + ev1 * ev1 + ev2 * ev2), 1e-10f);
  const float inv = 1.0f / nrm;
  const float evn[3] = {ev0 * inv, ev1 * inv, ev2 * inv};

  // filter weights for channels {t, 128+t, 256+t}
  float fw[3];
#pragma unroll
  for (int c = 0; c < 3; ++c) {
    const int j = c * NODE_SIZE + t;
    float acc = bf[j];
#pragma unroll
    for (int k = 0; k < EDGE_SIZE; ++k)
      acc = fmaf(es[k], Wf[k * (3 * NODE_SIZE) + j], acc);
    fw[c] = acc * cc;
  }

  // gather S[src], gate
  const float* __restrict__ srow = S + (size_t)src * (3 * NODE_SIZE);
  const float g_sv = fw[0] * srow[t];                   // gate_state_vector
  const float g_ev = fw[1] * srow[NODE_SIZE + t];       // gate_edge_vector
  const float msc  = fw[2] * srow[2 * NODE_SIZE + t];   // messages_scalar

  // scalar segment sum
  atomicAdd(&out_scalar[(size_t)dst * NODE_SIZE + t], msc);

  // vector messages + segment sum
  const float* __restrict__ vrow = nsv + (size_t)src * 3 * NODE_SIZE;
  float* __restrict__ orow = out_vector + (size_t)dst * 3 * NODE_SIZE;
#pragma unroll
  for (int a = 0; a < 3; ++a) {
    const float msg = fmaf(vrow[a * NODE_SIZE + t], g_sv, g_ev * evn[a]);
    atomicAdd(&orow[a * NODE_SIZE + t], msg);
  }
}

// ---------------------------------------------------------------------------
extern "C" void kernel_launch(void* const* d_in, const int* in_sizes, int n_in,
                              void* d_out, int out_size, void* d_ws, size_t ws_size,
                              hipStream_t stream) {
  const float* ns    = (const float*)d_in[0];   // node_state_scalar (20000,128)
  const float* nsv   = (const float*)d_in[1];   // node_state_vector (20000,3,128)
  const float* est   = (const float*)d_in[2];   // edge_state (320000,20)
  const float* ev    = (const float*)d_in[3];   // edge_vector (320000,3)
  const float* ed    = (const float*)d_in[4];   // edge_distance (320000,1)
  const int*   edges = (const int*)d_in[5];     // edges (320000,2)
  const float* Wf    = (const float*)d_in[6];
  const float* bf    = (const float*)d_in[7];
  const float* W1    = (const float*)d_in[8];
  const float* b1    = (const float*)d_in[9];
  const float* W2    = (const float*)d_in[10];
  const float* b2    = (const float*)d_in[11];

  float* out        = (float*)d_out;
  float* out_scalar = out;                                        // (N,128)
  float* out_vector = out + (size_t)NUM_NODES * NODE_SIZE;        // (N,3,128)
  float* S          = (float*)d_ws;    // 20000*384 f32 = 30.72 MB scratch

  // Pass 1: node MLP via f32 WMMA (1250 tiles of 16 nodes, 8 waves each).
  painn_node_mlp_kernel<<<NUM_NODES / 16, 256, 0, stream>>>(ns, W1, b1, W2, b2, S);

  // Pass 2: residual init of outputs.
  const size_t total = (size_t)NUM_NODES * NODE_SIZE * 4;
  painn_init_out_kernel<<<(unsigned)((total + 255) / 256), 256, 0, stream>>>(ns, nsv, out);

  // Pass 3: fused edge filter + gather + gate + atomic segment-sum.
  painn_edge_kernel<<<NUM_EDGES, 128, 0, stream>>>(est, ev, ed, edges, Wf, bf,
                                                   nsv, S, out_scalar, out_vector);
}